// RecurrentDCRNN_40037685133527
// MI455X (gfx1250) — compile-verified
//
#include <hip/hip_runtime.h>

typedef float v2f __attribute__((ext_vector_type(2)));
typedef float v8f __attribute__((ext_vector_type(8)));

namespace {
constexpr int N_NODES = 20000;
constexpr int N_EDGES = 320000;
constexpr int INC  = 64;    // in channels
constexpr int HIDC = 128;   // hidden
constexpr int OUTC = 64;    // out channels
constexpr int NTILE = N_NODES / 16;   // 1250 full 16-row tiles
constexpr int LDSTR = 132;            // padded LDS row stride (floats)

// workspace layout (float offsets)
constexpr int OFF_DEG_OUT  = 0;
constexpr int OFF_DEG_IN   = OFF_DEG_OUT + N_NODES;
constexpr int OFF_NORM_OUT = OFF_DEG_IN + N_NODES;
constexpr int OFF_NORM_IN  = OFF_NORM_OUT + N_EDGES;
constexpr int OFF_T1O      = OFF_NORM_IN + N_EDGES;
constexpr int OFF_T1I      = OFF_T1O + N_NODES * INC;
constexpr int OFF_BPACK    = OFF_T1I + N_NODES * INC;      // 6 * 64 * 128 floats
constexpr int OFF_LPACK    = OFF_BPACK + 6 * INC * HIDC;   // 128 * 64 floats
}

__device__ __forceinline__ v8f wmma4(v2f a, v2f b, v8f c) {
  // D = A(16x4,f32) * B(4x16,f32) + C(16x16,f32)
  return __builtin_amdgcn_wmma_f32_16x16x4_f32(false, a, false, b, (short)0, c,
                                               false, false);
}

// ---------------- zero init (degrees + T1 accumulators) ----------------
__global__ void k_init(float* __restrict__ ws) {
  const int total = 2 * N_NODES + 2 * N_NODES * INC;
  int i = blockIdx.x * blockDim.x + threadIdx.x;
  if (i >= total) return;
  if (i < 2 * N_NODES) ws[OFF_DEG_OUT + i] = 0.0f;
  else                 ws[OFF_T1O + (i - 2 * N_NODES)] = 0.0f;
}

// ---------------- degree accumulation ----------------
__global__ void k_degrees(const int* __restrict__ ei, const float* __restrict__ ew,
                          float* __restrict__ ws) {
  int e = blockIdx.x * blockDim.x + threadIdx.x;
  if (e >= N_EDGES) return;
  int r = ei[e];
  int c = ei[N_EDGES + e];
  float w = ew[e];
  atomicAdd(&ws[OFF_DEG_OUT + r], w);
  atomicAdd(&ws[OFF_DEG_IN + c], w);
}

// ---------------- per-edge normalized coefficients ----------------
__global__ void k_norms(const int* __restrict__ ei, const float* __restrict__ ew,
                        float* __restrict__ ws) {
  int e = blockIdx.x * blockDim.x + threadIdx.x;
  if (e >= N_EDGES) return;
  int r = ei[e];
  int c = ei[N_EDGES + e];
  float w = ew[e];
  float dob = ws[OFF_DEG_OUT + r];
  float dib = ws[OFF_DEG_IN + c];
  ws[OFF_NORM_OUT + e] = (dob > 0.0f) ? (w / dob) : 0.0f;
  ws[OFF_NORM_IN + e]  = (dib > 0.0f) ? (w / dib) : 0.0f;
}

// ---------------- edge scatter: T1o[col] += no*X[row]; T1i[row] += ni*X[col] ----------------
__global__ void k_scatter(const float* __restrict__ x, const int* __restrict__ ei,
                          float* __restrict__ ws) {
  int gid = blockIdx.x * blockDim.x + threadIdx.x;
  if (gid >= N_EDGES * 32) return;
  int e  = gid >> 5;
  int cp = (gid & 31) << 1;   // channel pair
  int r = ei[e];
  int c = ei[N_EDGES + e];
  float no = ws[OFF_NORM_OUT + e];
  float ni = ws[OFF_NORM_IN + e];
  v2f xr = *reinterpret_cast<const v2f*>(x + r * INC + cp);
  v2f xc = *reinterpret_cast<const v2f*>(x + c * INC + cp);
  atomicAdd(&ws[OFF_T1O + c * INC + cp + 0], no * xr[0]);
  atomicAdd(&ws[OFF_T1O + c * INC + cp + 1], no * xr[1]);
  atomicAdd(&ws[OFF_T1I + r * INC + cp + 0], ni * xc[0]);
  atomicAdd(&ws[OFF_T1I + r * INC + cp + 1], ni * xc[1]);
}

// ---------------- repack weights into WMMA B-fragment order ----------------
// B fragment for a 4x16 f32 tile: VGPR0 = {B[k0][n] lanes0-15, B[k0+2][n] lanes16-31},
// VGPR1 = {B[k0+1][n], B[k0+3][n]}.  We store the per-lane (v0,v1) pair contiguously:
// Bpack[((slot*8 + nt)*16 + kc)*64 + lane*2 + j],  k = 4*kc + 2*(lane>>4) + j,
// n = nt*16 + (lane&15).   Slots: 0..2 = z gate (W00+W10, W01, W11), 3..5 = h gate.
__global__ void k_pack(const float* __restrict__ Wz, const float* __restrict__ Wh,
                       const float* __restrict__ lw, float* __restrict__ ws) {
  const int NB = 6 * 8 * 16 * 32;   // v2 entries for gate weights
  const int NL = 4 * 32 * 32;       // v2 entries for lin_w
  int t = blockIdx.x * blockDim.x + threadIdx.x;
  if (t < NB) {
    int lane = t & 31;
    int kc   = (t >> 5) & 15;
    int nt   = (t >> 9) & 7;
    int s    = t >> 12;            // 0..5
    int n     = nt * 16 + (lane & 15);
    int kbase = kc * 4 + ((lane >> 4) << 1);
    const float* W = (s < 3) ? Wz : Wh;
    int sl = (s < 3) ? s : (s - 3);
    float v0, v1;
    if (sl == 0) {        // W[0,0] + W[1,0]  (direction sum, zeroth order)
      v0 = W[(kbase + 0) * HIDC + n] + W[2 * 192 * HIDC + (kbase + 0) * HIDC + n];
      v1 = W[(kbase + 1) * HIDC + n] + W[2 * 192 * HIDC + (kbase + 1) * HIDC + n];
    } else if (sl == 1) { // W[0,1] (forward first-order)
      v0 = W[1 * 192 * HIDC + (kbase + 0) * HIDC + n];
      v1 = W[1 * 192 * HIDC + (kbase + 1) * HIDC + n];
    } else {              // W[1,1] (reverse first-order)
      v0 = W[3 * 192 * HIDC + (kbase + 0) * HIDC + n];
      v1 = W[3 * 192 * HIDC + (kbase + 1) * HIDC + n];
    }
    ws[OFF_BPACK + t * 2 + 0] = v0;
    ws[OFF_BPACK + t * 2 + 1] = v1;
  } else if (t < NB + NL) {
    int u = t - NB;
    int lane = u & 31;
    int kc   = (u >> 5) & 31;
    int ot   = u >> 10;           // 0..3
    int n     = ot * 16 + (lane & 15);
    int kbase = kc * 4 + ((lane >> 4) << 1);
    ws[OFF_LPACK + u * 2 + 0] = lw[(kbase + 0) * OUTC + n];
    ws[OFF_LPACK + u * 2 + 1] = lw[(kbase + 1) * OUTC + n];
  }
}

// ---------------- fused WMMA: gates + GRU combine + head GEMM ----------------
__global__ __launch_bounds__(128)
void k_dcrnn(const float* __restrict__ x, const float* __restrict__ ws,
             const float* __restrict__ b_z, const float* __restrict__ b_h,
             const float* __restrict__ lin_b, float* __restrict__ out) {
  __shared__ float lds[4 * 16 * LDSTR];
  const int wave = threadIdx.x >> 5;
  const int lane = threadIdx.x & 31;
  const int tile = blockIdx.x * 4 + wave;
  if (tile >= NTILE) return;             // wave-uniform; no block barrier used

  const int m  = lane & 15;              // row inside 16-row tile
  const int kb = (lane >> 4) << 1;       // 0 or 2: K sub-column of A fragment
  const int r0 = tile * 16;

  const float* xr = x + (r0 + m) * INC;
  const float* to = ws + OFF_T1O + (r0 + m) * INC;
  const float* ti = ws + OFF_T1I + (r0 + m) * INC;

  // A fragments (16x4 f32 each), kept resident for the whole tile.
  v2f ax[16], ao[16], ai[16];
#pragma unroll
  for (int kc = 0; kc < 16; ++kc) {
    ax[kc] = *reinterpret_cast<const v2f*>(xr + kc * 4 + kb);
    ao[kc] = *reinterpret_cast<const v2f*>(to + kc * 4 + kb);
    ai[kc] = *reinterpret_cast<const v2f*>(ti + kc * 4 + kb);
  }

  const v2f* Bp = reinterpret_cast<const v2f*>(ws + OFF_BPACK);
  float* myLds = lds + wave * 16 * LDSTR;

  for (int nt = 0; nt < 8; ++nt) {       // 8 column tiles of the 128 hidden cols
    float bz = b_z[nt * 16 + m];
    float bh = b_h[nt * 16 + m];
    v8f az = {bz, bz, bz, bz, bz, bz, bz, bz};
    v8f ah = {bh, bh, bh, bh, bh, bh, bh, bh};
#pragma unroll
    for (int kc = 0; kc < 16; ++kc) {
      v2f bz0 = Bp[((0 * 8 + nt) * 16 + kc) * 32 + lane];
      v2f bz1 = Bp[((1 * 8 + nt) * 16 + kc) * 32 + lane];
      v2f bz2 = Bp[((2 * 8 + nt) * 16 + kc) * 32 + lane];
      az = wmma4(ax[kc], bz0, az);
      az = wmma4(ao[kc], bz1, az);
      az = wmma4(ai[kc], bz2, az);
      v2f bh0 = Bp[((3 * 8 + nt) * 16 + kc) * 32 + lane];
      v2f bh1 = Bp[((4 * 8 + nt) * 16 + kc) * 32 + lane];
      v2f bh2 = Bp[((5 * 8 + nt) * 16 + kc) * 32 + lane];
      ah = wmma4(ax[kc], bh0, ah);
      ah = wmma4(ao[kc], bh1, ah);
      ah = wmma4(ai[kc], bh2, ah);
    }
    // GRU combine with H0 = 0:  h = relu( sigmoid(-z_pre) * tanh(h_pre) )
#pragma unroll
    for (int r = 0; r < 8; ++r) {
      float oneMinusZ = 1.0f / (1.0f + __expf(az[r]));     // = 1 - sigmoid(z)
      float e2 = __expf(2.0f * ah[r]);
      float th = 1.0f - 2.0f / (e2 + 1.0f);                // tanh, Inf-safe
      float h = fmaxf(0.0f, oneMinusZ * th);
      int row = r + ((lane >> 4) << 3);
      myLds[row * LDSTR + nt * 16 + m] = h;
    }
  }

  // per-wave LDS transpose fence (split counter, CDNA5)
  asm volatile("s_wait_dscnt 0" ::: "memory");

  // reload h as A fragments (K = 128)
  v2f af[32];
#pragma unroll
  for (int kc = 0; kc < 32; ++kc)
    af[kc] = *reinterpret_cast<const v2f*>(myLds + m * LDSTR + kc * 4 + kb);

  const v2f* Lp = reinterpret_cast<const v2f*>(ws + OFF_LPACK);
  for (int ot = 0; ot < 4; ++ot) {       // 4 column tiles of the 64 outputs
    float lb = lin_b[ot * 16 + m];
    v8f acc = {lb, lb, lb, lb, lb, lb, lb, lb};
#pragma unroll
    for (int kc = 0; kc < 32; ++kc) {
      v2f b = Lp[(ot * 32 + kc) * 32 + lane];
      acc = wmma4(af[kc], b, acc);
    }
#pragma unroll
    for (int r = 0; r < 8; ++r) {
      int row = r0 + r + ((lane >> 4) << 3);
      out[row * OUTC + ot * 16 + m] = acc[r];
    }
  }
}

extern "C" void kernel_launch(void* const* d_in, const int* in_sizes, int n_in,
                              void* d_out, int out_size, void* d_ws, size_t ws_size,
                              hipStream_t stream) {
  const float* x   = (const float*)d_in[0];
  const int*   ei  = (const int*)d_in[1];
  const float* ew  = (const float*)d_in[2];
  const float* Wz  = (const float*)d_in[3];
  const float* bz  = (const float*)d_in[4];
  // d_in[5] = W_r, d_in[6] = b_r : dead (H0 == 0 -> R*H0 == 0, XRH == XH)
  const float* Wh  = (const float*)d_in[7];
  const float* bh  = (const float*)d_in[8];
  const float* lw  = (const float*)d_in[9];
  const float* lb  = (const float*)d_in[10];
  float* out = (float*)d_out;
  float* ws  = (float*)d_ws;

  (void)in_sizes; (void)n_in; (void)out_size; (void)ws_size;

  {
    const int total = 2 * N_NODES + 2 * N_NODES * INC;
    k_init<<<(total + 255) / 256, 256, 0, stream>>>(ws);
  }
  k_degrees<<<(N_EDGES + 255) / 256, 256, 0, stream>>>(ei, ew, ws);
  k_norms<<<(N_EDGES + 255) / 256, 256, 0, stream>>>(ei, ew, ws);
  k_scatter<<<(N_EDGES * 32 + 255) / 256, 256, 0, stream>>>(x, ei, ws);
  {
    const int total = 6 * 8 * 16 * 32 + 4 * 32 * 32;
    k_pack<<<(total + 255) / 256, 256, 0, stream>>>(Wz, Wh, lw, ws);
  }
  k_dcrnn<<<(NTILE + 3) / 4, 128, 0, stream>>>(x, ws, bz, bh, lb, out);
}